// TextGNN_7052336300299
// MI455X (gfx1250) — compile-verified
//
#include <hip/hip_runtime.h>
#include <hip/hip_bf16.h>
#include <math.h>

// Problem constants (match reference)
#define BB 16
#define TT 96
#define DD 768
#define VV 4096

typedef __attribute__((ext_vector_type(2))) float v2f;
typedef __attribute__((ext_vector_type(8))) float v8f;

// ---------------------------------------------------------------------------
// Kernel 1: w[b,s,t] = edge_weights[ edge_matrix[ tok[b,s], tok[b,t] ] ]
// One thread per (b,s,t). 147456 elements. edge_matrix is 67MB -> L2-resident.
// ---------------------------------------------------------------------------
__global__ void tg_w_kernel(const int* __restrict__ token_ids,
                            const float* __restrict__ edge_weights,
                            const int* __restrict__ edge_matrix,
                            float* __restrict__ w_buf) {
    int i = blockIdx.x * blockDim.x + threadIdx.x;
    const int total = BB * TT * TT;
    if (i >= total) return;
    int b = i / (TT * TT);
    int rem = i - b * (TT * TT);
    int s = rem / TT;
    int t = rem - s * TT;
    int ts = token_ids[b * TT + s];
    int td = token_ids[b * TT + t];
    int eid = edge_matrix[ts * VV + td];
    w_buf[i] = edge_weights[eid];   // layout: ((b*TT + s)*TT + t)
}

// ---------------------------------------------------------------------------
// Kernel 2: for each (b, d):  g[b,d] = mean_t( eta*H[b,t,d]
//                                            + (1-eta)*max_s H[b,s,d]*w[b,s,t] )
// Grid: b in [0,16) x 6 chunks of 128 d-columns. 128 threads, thread owns one d.
// H tile (96 x 128 f32 = 48KB) staged in LDS; w reads are block-uniform
// (scalar-load path). 96*96 = 9216 mul+max iterations per thread.
// ---------------------------------------------------------------------------
__global__ void tg_maxagg_kernel(const int* __restrict__ token_ids,
                                 const float* __restrict__ node_emb,
                                 const float* __restrict__ w_buf,
                                 const float* __restrict__ eta_p,
                                 float* __restrict__ g_buf) {
    __shared__ float Hs[TT * 128];   // 48 KB
    __shared__ int   toks[TT];

    const int chunk = blockIdx.x % (DD / 128);
    const int b     = blockIdx.x / (DD / 128);
    const int d0    = chunk * 128;
    const int tid   = threadIdx.x;   // 0..127

    if (tid < TT) toks[tid] = token_ids[b * TT + tid];
    __syncthreads();

    // Stage H[b, :, d0:d0+128] into LDS (coalesced 128-wide rows).
    for (int s = 0; s < TT; ++s) {
        Hs[s * 128 + tid] = node_emb[(size_t)toks[s] * DD + d0 + tid];
    }
    __syncthreads();

    const float eta = eta_p[0];
    const float one_m_eta = 1.0f - eta;
    const float* wb = w_buf + (size_t)b * TT * TT;   // wb[s*TT + t]

    float gacc = 0.0f;
    for (int t = 0; t < TT; ++t) {
        float acc = -INFINITY;
        #pragma unroll 4
        for (int s = 0; s < TT; ++s) {
            // w is uniform across the block for a given (s,t) -> scalar load.
            acc = fmaxf(acc, Hs[s * 128 + tid] * wb[s * TT + t]);
        }
        gacc += eta * Hs[t * 128 + tid] + one_m_eta * acc;
    }
    g_buf[b * DD + d0 + tid] = gacc * (1.0f / (float)TT);
}

// ---------------------------------------------------------------------------
// Kernel 3: out[b,n] = sigmoid( sum_k g[b,k]*W[n,k] + bias[n] )
// M=16 (one WMMA tile), N=768 (48 tiles), K=768 (192 steps of 4).
// One wave32 per 16x16 output tile; EXEC all-ones (WMMA requirement).
//
// V_WMMA_F32_16X16X4_F32 operand layout (ISA 7.12.2):
//   A (16x4): lanes 0-15 -> M=lane,  v0=K0 v1=K1 ; lanes 16-31 -> v0=K2 v1=K3
//   B (4x16): lanes 0-15 -> N=lane,  v0=K0 v1=K1 ; lanes 16-31 -> v0=K2 v1=K3
//   C/D (16x16): VGPR r, lanes 0-15 -> (M=r,   N=lane)
//                         lanes 16-31 -> (M=r+8, N=lane-16)
// ---------------------------------------------------------------------------
__global__ void tg_gemm_sigmoid_wmma(const float* __restrict__ g_buf,
                                     const float* __restrict__ Wm,
                                     const float* __restrict__ bias,
                                     float* __restrict__ out) {
    const int n0   = blockIdx.x * 16;
    const int lane = threadIdx.x;        // 0..31
    const int half = lane >> 4;          // 0: K pair {0,1}; 1: K pair {2,3}
    const int l    = lane & 15;          // row for A, col for B within tile

    v8f c = {};
    const float* arow = g_buf + (size_t)l * DD;          // A row M=l
    const float* brow = Wm    + (size_t)(n0 + l) * DD;   // B col N=n0+l (= W row)

    #pragma unroll 8
    for (int k0 = 0; k0 < DD; k0 += 4) {
        const int k = k0 + 2 * half;
        v2f a, bf;
        a.x  = arow[k + 0];
        a.y  = arow[k + 1];
        bf.x = brow[k + 0];
        bf.y = brow[k + 1];
        c = __builtin_amdgcn_wmma_f32_16x16x4_f32(
                /*neg_a=*/false, a, /*neg_b=*/false, bf,
                /*c_mod=*/(short)0, c, /*reuse_a=*/false, /*reuse_b=*/false);
    }

    const float bb = bias[n0 + l];
    #pragma unroll
    for (int r = 0; r < 8; ++r) {
        const int m = r + 8 * half;
        const float v = c[r] + bb;
        out[(size_t)m * DD + n0 + l] = 1.0f / (1.0f + expf(-v));
    }
}

// ---------------------------------------------------------------------------
// Launch wrapper
// inputs: 0 token_ids(i32 B*T) 1 node_emb(f32 V*D) 2 edge_weights(f32 E)
//         3 edge_matrix(i32 V*V) 4 eta(f32 1) 5 W(f32 D*D) 6 b(f32 D)
// out: f32 B*D
// ws: w_buf (B*T*T f32 = 589824 B) + g_buf (B*D f32 = 49152 B)
// ---------------------------------------------------------------------------
extern "C" void kernel_launch(void* const* d_in, const int* in_sizes, int n_in,
                              void* d_out, int out_size, void* d_ws, size_t ws_size,
                              hipStream_t stream) {
    const int*   token_ids    = (const int*)  d_in[0];
    const float* node_emb     = (const float*)d_in[1];
    const float* edge_weights = (const float*)d_in[2];
    const int*   edge_matrix  = (const int*)  d_in[3];
    const float* eta          = (const float*)d_in[4];
    const float* Wm           = (const float*)d_in[5];
    const float* bias         = (const float*)d_in[6];
    float*       out          = (float*)d_out;

    float* w_buf = (float*)d_ws;                       // B*T*T
    float* g_buf = w_buf + (size_t)BB * TT * TT;       // B*D

    // 1) edge-weight gather
    {
        const int total = BB * TT * TT;
        dim3 grid((total + 255) / 256), block(256);
        tg_w_kernel<<<grid, block, 0, stream>>>(token_ids, edge_weights,
                                                edge_matrix, w_buf);
    }
    // 2) max-times aggregation + eta blend + mean over t
    {
        dim3 grid(BB * (DD / 128)), block(128);
        tg_maxagg_kernel<<<grid, block, 0, stream>>>(token_ids, node_emb,
                                                     w_buf, eta, g_buf);
    }
    // 3) projection GEMM (f32 WMMA) + bias + sigmoid
    {
        dim3 grid(DD / 16), block(32);
        tg_gemm_sigmoid_wmma<<<grid, block, 0, stream>>>(g_buf, Wm, bias, out);
    }
}